// CustomModel_20667382628960
// MI455X (gfx1250) — compile-verified
//
#include <hip/hip_runtime.h>
#include <hip/hip_bf16.h>
#include <math.h>

// ---------------------------------------------------------------------------
// Problem constants
// ---------------------------------------------------------------------------
#define VOCAB 50265
#define HDIM  768
#define BS    16
#define LSEQ  256
#define LTT   128
#define LAT   64
#define CAPLEN 64
#define TOPK  5

typedef float v2f __attribute__((ext_vector_type(2)));
typedef float v8f __attribute__((ext_vector_type(8)));

__device__ __forceinline__ v8f wmma_f32(v2f a, v2f b, v8f c) {
    // D = A(16x4) * B(4x16) + C(16x16), all f32, wave32.
    return __builtin_amdgcn_wmma_f32_16x16x4_f32(
        /*neg_a=*/false, a, /*neg_b=*/false, b,
        /*c_mod=*/(short)0, c, /*reuse_a=*/false, /*reuse_b=*/false);
}

__device__ __forceinline__ bool better(float v1, int i1, float v2, int i2) {
    // jax.lax.top_k: descending values, ties broken by lower index first.
    return (v1 > v2) || (v1 == v2 && i1 < i2);
}

// ---------------------------------------------------------------------------
// Kernel 1: per-batch length reductions
// lens[0..15]  = origin_len, [16..31] = tt_len, [32..47] = target_len,
// lens[48..63] = caption_len (zeros of cap_mask|fin_hot)
// ---------------------------------------------------------------------------
__global__ void lengths_kernel(const int* __restrict__ am,
                               const int* __restrict__ amtt,
                               const int* __restrict__ amat,
                               const unsigned char* __restrict__ cap_mask,
                               const int* __restrict__ finished,
                               int* __restrict__ lens) {
    __shared__ int red[256];
    const int b = blockIdx.x;
    const int tid = threadIdx.x;

    red[tid] = am[b * LSEQ + tid];
    __syncthreads();
    for (int off = 128; off > 0; off >>= 1) {
        if (tid < off) red[tid] += red[tid + off];
        __syncthreads();
    }
    if (tid == 0) lens[b] = red[0];
    __syncthreads();

    red[tid] = (tid < LTT) ? amtt[b * LTT + tid] : 0;
    __syncthreads();
    for (int off = 128; off > 0; off >>= 1) {
        if (tid < off) red[tid] += red[tid + off];
        __syncthreads();
    }
    if (tid == 0) lens[16 + b] = red[0];
    __syncthreads();

    red[tid] = (tid < LAT) ? amat[b * LAT + tid] : 0;
    __syncthreads();
    for (int off = 128; off > 0; off >>= 1) {
        if (tid < off) red[tid] += red[tid + off];
        __syncthreads();
    }
    if (tid == 0) lens[32 + b] = red[0];
    __syncthreads();

    int freec = 0;
    if (tid < CAPLEN) {
        const int fin = finished[b];
        const bool fh = (tid == fin) && (fin > 0);
        const bool cm = cap_mask[b * CAPLEN + tid] != 0;
        freec = (cm || fh) ? 0 : 1;
    }
    red[tid] = freec;
    __syncthreads();
    for (int off = 128; off > 0; off >>= 1) {
        if (tid < off) red[tid] += red[tid + off];
        __syncthreads();
    }
    if (tid == 0) lens[48 + b] = red[0];
}

// ---------------------------------------------------------------------------
// Kernel 2: streaming top-5 over VOCAB per (b, cap) row, softmax(top5),
// weighted 5-row gather of W_emb -> sce[row][0..767].
// Main loop processes 4 strided elements per iteration (4 loads in flight,
// one long-distance prefetch per group).  50265 = 49*1024 + 89.
// ---------------------------------------------------------------------------
__global__ void topk_sce_kernel(const float* __restrict__ caption_out,
                                const float* __restrict__ W_emb,
                                float* __restrict__ sce) {
    const int row = blockIdx.x;   // b*CAPLEN + c
    const int tid = threadIdx.x;  // 0..255
    const float* __restrict__ src = caption_out + (size_t)row * VOCAB;

    float tv[TOPK];
    int   ti[TOPK];
#pragma unroll
    for (int k = 0; k < TOPK; ++k) { tv[k] = -INFINITY; ti[k] = 0x7fffffff; }

#define TK_INSERT(vv, ii)                                                     \
    do {                                                                      \
        if (better((vv), (ii), tv[TOPK - 1], ti[TOPK - 1])) {                 \
            tv[TOPK - 1] = (vv);                                              \
            ti[TOPK - 1] = (ii);                                              \
            _Pragma("unroll")                                                 \
            for (int k = TOPK - 1; k > 0; --k) {                              \
                if (better(tv[k], ti[k], tv[k - 1], ti[k - 1])) {             \
                    const float fv = tv[k]; tv[k] = tv[k - 1]; tv[k - 1] = fv;\
                    const int iv = ti[k]; ti[k] = ti[k - 1]; ti[k - 1] = iv;  \
                }                                                             \
            }                                                                 \
        }                                                                     \
    } while (0)

    const int vmain = (VOCAB / 1024) * 1024;   // 50176
    for (int base = 0; base < vmain; base += 1024) {
        const int i0 = base + tid;
        if (i0 + 8192 < VOCAB) __builtin_prefetch(&src[i0 + 8192], 0, 1);
        const float v0 = src[i0];
        const float v1 = src[i0 + 256];
        const float v2 = src[i0 + 512];
        const float v3 = src[i0 + 768];
        TK_INSERT(v0, i0);
        TK_INSERT(v1, i0 + 256);
        TK_INSERT(v2, i0 + 512);
        TK_INSERT(v3, i0 + 768);
    }
    for (int i = vmain + tid; i < VOCAB; i += 256) {
        const float v = src[i];
        TK_INSERT(v, i);
    }
#undef TK_INSERT

    __shared__ float sv[256 * TOPK];
    __shared__ int   si[256 * TOPK];
#pragma unroll
    for (int k = 0; k < TOPK; ++k) { sv[tid * TOPK + k] = tv[k]; si[tid * TOPK + k] = ti[k]; }
    __syncthreads();

    // tree-merge sorted-5 lists
    for (int off = 128; off > 0; off >>= 1) {
        if (tid < off) {
            const int ba = tid * TOPK;
            const int bb = (tid + off) * TOPK;
            int ia = 0, ib = 0;
            float ov[TOPK]; int oi[TOPK];
#pragma unroll
            for (int k = 0; k < TOPK; ++k) {
                const float va = sv[ba + ia]; const int xa = si[ba + ia];
                const float vb = sv[bb + ib]; const int xb = si[bb + ib];
                if (better(va, xa, vb, xb)) { ov[k] = va; oi[k] = xa; ++ia; }
                else                        { ov[k] = vb; oi[k] = xb; ++ib; }
            }
#pragma unroll
            for (int k = 0; k < TOPK; ++k) { sv[ba + k] = ov[k]; si[ba + k] = oi[k]; }
        }
        __syncthreads();
    }

    __shared__ float pk[TOPK];
    __shared__ int   ik[TOPK];
    if (tid == 0) {
        const float m = sv[0];  // sorted desc -> max first
        float e[TOPK], s = 0.0f;
#pragma unroll
        for (int k = 0; k < TOPK; ++k) { e[k] = __expf(sv[k] - m); s += e[k]; }
        const float inv = 1.0f / s;
#pragma unroll
        for (int k = 0; k < TOPK; ++k) { pk[k] = e[k] * inv; ik[k] = si[k]; }
    }
    __syncthreads();

    const float p0 = pk[0], p1 = pk[1], p2 = pk[2], p3 = pk[3], p4 = pk[4];
    const float* __restrict__ w0 = W_emb + (size_t)ik[0] * HDIM;
    const float* __restrict__ w1 = W_emb + (size_t)ik[1] * HDIM;
    const float* __restrict__ w2 = W_emb + (size_t)ik[2] * HDIM;
    const float* __restrict__ w3 = W_emb + (size_t)ik[3] * HDIM;
    const float* __restrict__ w4 = W_emb + (size_t)ik[4] * HDIM;
    float* __restrict__ dst = sce + (size_t)row * HDIM;
    for (int h = tid; h < HDIM; h += 256) {
        dst[h] = p0 * w0[h] + p1 * w1[h] + p2 * w2[h] + p3 * w3[h] + p4 * w4[h];
    }
}

// ---------------------------------------------------------------------------
// Kernel 3: row gather  dst[row] = W_emb[ids[row]]   (block = 192 thr, float4)
// ---------------------------------------------------------------------------
__global__ void gather_rows_kernel(float* __restrict__ dst,
                                   const float* __restrict__ W,
                                   const int* __restrict__ ids) {
    const int row = blockIdx.x;
    const float4* __restrict__ s = (const float4*)(W + (size_t)ids[row] * HDIM);
    float4* __restrict__ d = (float4*)(dst + (size_t)row * HDIM);
    d[threadIdx.x] = s[threadIdx.x];
}

// ---------------------------------------------------------------------------
// Kernel 4: batched attention with f32 WMMA (16x16x4).
//   out = softmax(Q K^T / sqrt(768), key-masked by klen) @ KV
//   if resid != nullptr: out = 0.5*(resid + out)
// One wave (32 threads) per 16-row Q strip; grid = BS * (64/16).
// B fragments are batched into register arrays so the scheduler can keep
// many global loads in flight instead of load->wait0->wmma serialization.
// ---------------------------------------------------------------------------
template <int LK>
__global__ void attn_kernel(const float* __restrict__ Q,
                            const float* __restrict__ KV,
                            const int* __restrict__ klen,
                            const float* __restrict__ resid,
                            float* __restrict__ out) {
    constexpr int NT = LK / 16;
    __shared__ float P[16 * LK];

    const int b    = blockIdx.x >> 2;
    const int m0   = (blockIdx.x & 3) << 4;
    const int lane = threadIdx.x;       // 0..31, all active (EXEC all ones)
    const int half = lane >> 4;
    const int lr   = lane & 15;

    const float* __restrict__ Qb = Q  + (size_t)b * LAT * HDIM;
    const float* __restrict__ Kb = KV + (size_t)b * LK * HDIM;
    const int kl = klen[b];

    v8f acc[NT];
#pragma unroll
    for (int t = 0; t < NT; ++t)
#pragma unroll
        for (int j = 0; j < 8; ++j) acc[t][j] = 0.0f;

    // ---- scores: Q (16 x 768) * K^T (768 x LK), K-step = 4, unroll x2 ------
    const float* __restrict__ qrow = Qb + (size_t)(m0 + lr) * HDIM + 2 * half;
    const float* __restrict__ krow = Kb + (size_t)lr * HDIM + 2 * half;
    for (int kk = 0; kk < HDIM; kk += 8) {
        const v2f a0 = *(const v2f*)(qrow + kk);
        const v2f a1 = *(const v2f*)(qrow + kk + 4);
        v2f bf0[NT], bf1[NT];
#pragma unroll
        for (int t = 0; t < NT; ++t)
            bf0[t] = *(const v2f*)(krow + (size_t)t * 16 * HDIM + kk);
#pragma unroll
        for (int t = 0; t < NT; ++t)
            bf1[t] = *(const v2f*)(krow + (size_t)t * 16 * HDIM + kk + 4);
#pragma unroll
        for (int t = 0; t < NT; ++t)
            acc[t] = wmma_f32(a0, bf0[t], acc[t]);
#pragma unroll
        for (int t = 0; t < NT; ++t)
            acc[t] = wmma_f32(a1, bf1[t], acc[t]);
    }

    // ---- scale + key mask (matches reference: mask AFTER scaling) ----------
    const float scale = 0.03608439182435161f;   // 1/sqrt(768)
#pragma unroll
    for (int t = 0; t < NT; ++t) {
        const bool valid = (t * 16 + lr) < kl;  // column index == lr per half
#pragma unroll
        for (int j = 0; j < 8; ++j) {
            const float v = acc[t][j] * scale;
            acc[t][j] = valid ? v : -1.0e9f;
        }
    }

    // ---- row softmax: row (j + 8*half) lives on one 16-lane half -----------
#pragma unroll
    for (int j = 0; j < 8; ++j) {
        float rm = -INFINITY;
#pragma unroll
        for (int t = 0; t < NT; ++t) rm = fmaxf(rm, acc[t][j]);
#pragma unroll
        for (int off = 8; off >= 1; off >>= 1)
            rm = fmaxf(rm, __shfl_xor(rm, off, 16));
        float rs = 0.0f;
#pragma unroll
        for (int t = 0; t < NT; ++t) {
            const float e = __expf(acc[t][j] - rm);
            acc[t][j] = e;
            rs += e;
        }
#pragma unroll
        for (int off = 8; off >= 1; off >>= 1)
            rs += __shfl_xor(rs, off, 16);
        const float inv = 1.0f / rs;
        const int prow = j + 8 * half;
#pragma unroll
        for (int t = 0; t < NT; ++t)
            P[prow * LK + t * 16 + lr] = acc[t][j] * inv;
    }
    __syncthreads();

    // ---- out strip: P (16 x LK) @ V (LK x 768), K-step 8, batched loads ----
    const float* __restrict__ prow_lds = &P[lr * LK + 2 * half];
    for (int n0 = 0; n0 < HDIM; n0 += 16) {
        const float* __restrict__ vcol = Kb + (size_t)2 * half * HDIM + n0 + lr;
        v8f o;
#pragma unroll
        for (int j = 0; j < 8; ++j) o[j] = 0.0f;
#pragma unroll 2
        for (int kk = 0; kk < LK; kk += 8) {
            const v2f a0 = *(const v2f*)(prow_lds + kk);
            const v2f a1 = *(const v2f*)(prow_lds + kk + 4);
            const float b00 = vcol[(size_t)kk * HDIM];
            const float b01 = vcol[(size_t)(kk + 1) * HDIM];
            const float b10 = vcol[(size_t)(kk + 4) * HDIM];
            const float b11 = vcol[(size_t)(kk + 5) * HDIM];
            v2f bf0; bf0.x = b00; bf0.y = b01;
            v2f bf1; bf1.x = b10; bf1.y = b11;
            o = wmma_f32(a0, bf0, o);
            o = wmma_f32(a1, bf1, o);
        }
#pragma unroll
        for (int j = 0; j < 8; ++j) {
            const int row = m0 + j + 8 * half;
            const size_t idx = (size_t)b * LAT * HDIM + (size_t)row * HDIM + n0 + lr;
            float v = o[j];
            if (resid) v = 0.5f * (resid[idx] + v);
            out[idx] = v;
        }
    }
}

// ---------------------------------------------------------------------------
// Kernel 5: final compose of inputs_embeds (BS,L,H) + new_mask (BS,L as f32)
// ---------------------------------------------------------------------------
__global__ void finalize_kernel(const float* __restrict__ att_ai,
                                const float* __restrict__ W,
                                const int* __restrict__ input_ids,
                                const int* __restrict__ attention_mask,
                                const int* __restrict__ lens,
                                const int* __restrict__ end_token_p,
                                float* __restrict__ out) {
    const int row = blockIdx.x;          // b*LSEQ + p
    const int b = row >> 8;
    const int p = row & 255;
    const int o = lens[b];
    const int t = lens[32 + b];
    const bool in_range = (p >= o) && (p < o + t);
    const bool is_end   = (p == o + t);

    const float* __restrict__ src;
    if (in_range) {
        int rel = p - o;
        if (rel > LAT - 1) rel = LAT - 1;
        src = att_ai + ((size_t)b * LAT + rel) * HDIM;
    } else if (is_end) {
        src = W + (size_t)end_token_p[0] * HDIM;
    } else {
        src = W + (size_t)input_ids[row] * HDIM;
    }
    const float4* __restrict__ s4 = (const float4*)src;
    float4* __restrict__ d4 = (float4*)(out + (size_t)row * HDIM);
    d4[threadIdx.x] = s4[threadIdx.x];

    if (threadIdx.x == 0) {
        out[(size_t)BS * LSEQ * HDIM + row] =
            in_range ? 1.0f : (float)attention_mask[row];
    }
}

// ---------------------------------------------------------------------------
// Host launcher
// ---------------------------------------------------------------------------
extern "C" void kernel_launch(void* const* d_in, const int* in_sizes, int n_in,
                              void* d_out, int out_size, void* d_ws, size_t ws_size,
                              hipStream_t stream) {
    const float* caption_out          = (const float*)d_in[0];
    const unsigned char* cap_mask     = (const unsigned char*)d_in[1]; // bool array
    const int* finished               = (const int*)d_in[2];
    const int* input_ids              = (const int*)d_in[3];
    const int* attention_mask         = (const int*)d_in[4];
    const int* input_ids_tt           = (const int*)d_in[5];
    const int* attention_mask_tt      = (const int*)d_in[6];
    const int* input_ids_at           = (const int*)d_in[7];
    const int* attention_mask_at      = (const int*)d_in[8];
    const float* W_emb                = (const float*)d_in[9];
    const int* end_token_p            = (const int*)d_in[10];
    float* outf                       = (float*)d_out;

    // workspace layout
    float* sce    = (float*)d_ws;                        // 16*64*768
    float* emb_tt = sce    + (size_t)BS * CAPLEN * HDIM; // 16*128*768
    float* emb_at = emb_tt + (size_t)BS * LTT * HDIM;    // 16*64*768
    float* qp     = emb_at + (size_t)BS * LAT * HDIM;    // 16*64*768
    float* att_ai = qp     + (size_t)BS * LAT * HDIM;    // 16*64*768
    int*   lens   = (int*)(att_ai + (size_t)BS * LAT * HDIM); // 64 ints

    lengths_kernel<<<BS, 256, 0, stream>>>(attention_mask, attention_mask_tt,
                                           attention_mask_at, cap_mask, finished, lens);

    topk_sce_kernel<<<BS * CAPLEN, 256, 0, stream>>>(caption_out, W_emb, sce);

    gather_rows_kernel<<<BS * LTT, HDIM / 4, 0, stream>>>(emb_tt, W_emb, input_ids_tt);
    gather_rows_kernel<<<BS * LAT, HDIM / 4, 0, stream>>>(emb_at, W_emb, input_ids_at);

    // attention 1: emb_at' = 0.5*(emb_at + attn(emb_at, emb_tt, emb_tt, tt_len))
    attn_kernel<LTT><<<BS * (LAT / 16), 32, 0, stream>>>(emb_at, emb_tt,
                                                         lens + 16, emb_at, qp);
    // attention 2: att_ai = attn(emb_at', sce, sce, caption_len)
    attn_kernel<CAPLEN><<<BS * (LAT / 16), 32, 0, stream>>>(qp, sce,
                                                            lens + 48, nullptr, att_ai);

    finalize_kernel<<<BS * LSEQ, HDIM / 4, 0, stream>>>(att_ai, W_emb, input_ids,
                                                        attention_mask, lens,
                                                        end_token_p, outf);
}